// UnPatcher_19052474925282
// MI455X (gfx1250) — compile-verified
//
#include <hip/hip_runtime.h>
#include <stdint.h>

typedef float f32x4 __attribute__((ext_vector_type(4)));

namespace {
constexpr int Bd = 2, Gd = 16, Td = 8, Hd = 128, Wd = 128;
constexpr int SUB_STRIDE = Gd * Td * Hd * Wd;       // 2,097,152 elems between subbands
constexpr int ROWS = Bd * Gd * Td * Hd;             // 32,768 (b,g,t,h) rows
constexpr int WAVES_PER_BLOCK = 8;                  // 256 threads = 8 wave32
constexpr int BLOCKS = 512;
constexpr int N_WAVES = BLOCKS * WAVES_PER_BLOCK;   // 4096
constexpr int ITERS = ROWS / N_WAVES;               // 8
constexpr int OUT_T = 2 * Td, OUT_H = 2 * Hd, OUT_W = 2 * Wd;
}

// Async global -> LDS copy, 16B per lane (one full 512B row per wave-issue).
// GV addressing: 64-bit VGPR address, LDS byte offset in a VGPR. Tracked by ASYNCcnt.
__device__ __forceinline__ void async_row_load(unsigned lds_byte, const float* g) {
  asm volatile("global_load_async_to_lds_b128 %0, %1, off"
               :: "v"(lds_byte), "v"((unsigned long long)(uintptr_t)g)
               : "memory");
}

__global__ __launch_bounds__(WAVES_PER_BLOCK * 32)
void ihaar3d_async_kernel(const float* __restrict__ x, float* __restrict__ out) {
  // [wave][double-buffer slot][subband][W]: 8KB per wave, 64KB per block.
  __shared__ __align__(16) float buf[WAVES_PER_BLOCK][2][8][Wd];

  const int lane = threadIdx.x & 31;
  const int warp = threadIdx.x >> 5;
  const int wid  = blockIdx.x * WAVES_PER_BLOCK + warp;

  // Issue the 8 subband async loads of row `it*N_WAVES+wid` into `slot`.
  auto issue = [&](int it, int slot) {
    const int row = it * N_WAVES + wid;
    const int h = row & (Hd - 1);
    const int t = (row >> 7) & (Td - 1);
    const int g = (row >> 10) & (Gd - 1);
    const int b = row >> 14;
    // channel = s*16 + g ; x index = (((b*128 + ch)*T + t)*H + h)*W + w
    const int base0 = (((b * 128 + g) * Td + t) * Hd + h) * Wd + lane * 4;
    // low 32 bits of a flat shared address == LDS byte offset (aperture in [63:32])
    const unsigned l0 = (unsigned)(uintptr_t)(void*)&buf[warp][slot][0][lane * 4];
#pragma unroll
    for (int s = 0; s < 8; ++s)
      async_row_load(l0 + s * (Wd * 4), x + base0 + s * SUB_STRIDE);
  };

  issue(0, 0);
  int slot = 0;
#pragma unroll 1
  for (int it = 0; it < ITERS; ++it) {
    if (it + 1 < ITERS) {
      issue(it + 1, slot ^ 1);                      // prefetch next row set
      asm volatile("s_wait_asynccnt 8" ::: "memory"); // current 8 loads done
    } else {
      asm volatile("s_wait_asynccnt 0" ::: "memory");
    }

    // Each lane pulls its 4 consecutive w values for all 8 subbands (ds_load_b128).
    f32x4 A[8];
#pragma unroll
    for (int s = 0; s < 8; ++s)
      A[s] = *(const f32x4*)&buf[warp][slot][s][lane * 4];

    // 3-stage Hadamard butterfly per w element.
    // subband s = bt*4 + bh*2 + bw ; sign = (-1)^(bt*dt + bh*dh + bw*dw)
    float ov[2][2][8];  // [dt][dh][8 consecutive output w]
#pragma unroll
    for (int k = 0; k < 4; ++k) {
      const float s0 = A[0][k], s1 = A[1][k], s2 = A[2][k], s3 = A[3][k];
      const float s4 = A[4][k], s5 = A[5][k], s6 = A[6][k], s7 = A[7][k];
      // W stage (bw): p -> dw=0 (even), m -> dw=1 (odd)
      const float w0p = s0 + s1, w0m = s0 - s1;
      const float w1p = s2 + s3, w1m = s2 - s3;
      const float w2p = s4 + s5, w2m = s4 - s5;
      const float w3p = s6 + s7, w3m = s6 - s7;
      // H stage (bh): h** = dw0 branch, g** = dw1 branch; second index = dh
      const float h00 = w0p + w1p, h01 = w0p - w1p;
      const float g00 = w0m + w1m, g01 = w0m - w1m;
      const float h10 = w2p + w3p, h11 = w2p - w3p;
      const float g10 = w2m + w3m, g11 = w2m - w3m;
      // T stage (bt): dt=0 -> sum, dt=1 -> diff. Interleave even/odd along W.
      ov[0][0][2 * k] = h00 + h10;  ov[0][0][2 * k + 1] = g00 + g10;
      ov[0][1][2 * k] = h01 + h11;  ov[0][1][2 * k + 1] = g01 + g11;
      ov[1][0][2 * k] = h00 - h10;  ov[1][0][2 * k + 1] = g00 - g10;
      ov[1][1][2 * k] = h01 - h11;  ov[1][1][2 * k + 1] = g01 - g11;
    }

    const int row = it * N_WAVES + wid;
    const int h = row & (Hd - 1);
    const int t = (row >> 7) & (Td - 1);
    const int g = (row >> 10) & (Gd - 1);
    const int b = row >> 14;

    // 4 output rows x 8 floats = 2x float4 NT stores each (write-once stream).
#pragma unroll
    for (int dt = 0; dt < 2; ++dt) {
#pragma unroll
      for (int dh = 0; dh < 2; ++dh) {
        const long long rb =
            ((long long)((b * Gd + g) * OUT_T + 2 * t + dt) * OUT_H + (2 * h + dh))
                * OUT_W + 8 * lane;
        f32x4 v0 = { ov[dt][dh][0], ov[dt][dh][1], ov[dt][dh][2], ov[dt][dh][3] };
        f32x4 v1 = { ov[dt][dh][4], ov[dt][dh][5], ov[dt][dh][6], ov[dt][dh][7] };
        v0 *= 0.125f;
        v1 *= 0.125f;
        __builtin_nontemporal_store(v0, (f32x4*)(out + rb));
        __builtin_nontemporal_store(v1, (f32x4*)(out + rb + 4));
      }
    }

    slot ^= 1;
  }
}

extern "C" void kernel_launch(void* const* d_in, const int* in_sizes, int n_in,
                              void* d_out, int out_size, void* d_ws, size_t ws_size,
                              hipStream_t stream) {
  (void)in_sizes; (void)n_in; (void)d_ws; (void)ws_size; (void)out_size;
  const float* x = (const float*)d_in[0];
  float* out = (float*)d_out;
  ihaar3d_async_kernel<<<dim3(BLOCKS), dim3(WAVES_PER_BLOCK * 32), 0, stream>>>(x, out);
}